// HierarchicalCrossEntropy_14877766713481
// MI455X (gfx1250) — compile-verified
//
#include <hip/hip_runtime.h>
#include <stdint.h>

// Hierarchical cross-entropy on MI455X (gfx1250).
// Bandwidth-bound streaming reduction: 128 MB of logits @ 23.3 TB/s ~= 5.5us floor.
// Uses CDNA5 async global->LDS copies (ASYNCcnt) with double buffering per wave.

#define C_CLASSES 1024
#define WAVES_PER_BLOCK 8
#define ROWS_PER_WAVE 4
#define ROWS_PER_BLOCK (WAVES_PER_BLOCK * ROWS_PER_WAVE)

// Issue 8x b128 async copies: one full 4KB row (1024 fp32) into an LDS buffer.
// Per CDNA5 ISA: INST_OFFSET is added to BOTH the LDS address
// (VDST vgpr) and the global address (SADDR + VADDR vgpr), and our LDS/global
// strides per iteration are both 512B, so one immediate serves both sides.
__device__ __forceinline__ void issue_row_async(uint32_t ldsAddr, uint32_t gOff,
                                                const float* __restrict__ gbase) {
    asm volatile(
        "global_load_async_to_lds_b128 %0, %1, %2 offset:0\n\t"
        "global_load_async_to_lds_b128 %0, %1, %2 offset:512\n\t"
        "global_load_async_to_lds_b128 %0, %1, %2 offset:1024\n\t"
        "global_load_async_to_lds_b128 %0, %1, %2 offset:1536\n\t"
        "global_load_async_to_lds_b128 %0, %1, %2 offset:2048\n\t"
        "global_load_async_to_lds_b128 %0, %1, %2 offset:2560\n\t"
        "global_load_async_to_lds_b128 %0, %1, %2 offset:3072\n\t"
        "global_load_async_to_lds_b128 %0, %1, %2 offset:3584"
        :: "v"(ldsAddr), "v"(gOff), "s"(gbase) : "memory");
}

__device__ __forceinline__ void wait_async_le8() {
    asm volatile("s_wait_asynccnt 8" ::: "memory");
}
__device__ __forceinline__ void wait_async_0() {
    asm volatile("s_wait_asynccnt 0" ::: "memory");
}

__device__ __forceinline__ float wred_sum(float v) {
    #pragma unroll
    for (int m = 16; m > 0; m >>= 1) v += __shfl_xor(v, m, 32);
    return v;
}
__device__ __forceinline__ float wred_max(float v) {
    #pragma unroll
    for (int m = 16; m > 0; m >>= 1) v = fmaxf(v, __shfl_xor(v, m, 32));
    return v;
}

__global__ __launch_bounds__(256) void hce_rows_kernel(
        const float* __restrict__ logits,   // (B, 1024)
        const int*   __restrict__ tgt,      // (B,)
        const float* __restrict__ lam,      // (3,)
        float*       __restrict__ partial,  // (gridDim.x,)
        int Btot) {
    // 8 waves * 2 buffers * 4KB = 64KB LDS (WGP has 320KB).
    __shared__ alignas(16) float lds[WAVES_PER_BLOCK][2][C_CLASSES];
    __shared__ float wsum[WAVES_PER_BLOCK];

    const int wave = threadIdx.x >> 5;
    const int lane = threadIdx.x & 31;

    const float lam0 = lam[0], lam1 = lam[1], lam2 = lam[2];

    // Low 32 bits of the flat address of a __shared__ object == LDS byte offset
    // (aperture lives in the high bits; DS/async HW adds LDS_BASE itself).
    uint32_t ldsA[2];
    ldsA[0] = (uint32_t)(uintptr_t)(&lds[wave][0][0]) + (uint32_t)lane * 16u;
    ldsA[1] = (uint32_t)(uintptr_t)(&lds[wave][1][0]) + (uint32_t)lane * 16u;

    const int row0 = ((int)blockIdx.x * WAVES_PER_BLOCK + wave) * ROWS_PER_WAVE;
    auto clampRow = [&](int r) { return r < Btot ? r : (Btot - 1); };

    // Prime the pipeline: row 0 of this wave into buffer 0.
    issue_row_async(ldsA[0], (uint32_t)clampRow(row0) * 4096u + (uint32_t)lane * 16u, logits);

    float wloss = 0.0f;

    for (int j = 0; j < ROWS_PER_WAVE; ++j) {
        const int r = row0 + j;
        if (j + 1 < ROWS_PER_WAVE) {
            // Prefetch next row into the other buffer, then wait for the
            // oldest 8 copies (this row) to land: async loads complete in order.
            issue_row_async(ldsA[(j + 1) & 1],
                            (uint32_t)clampRow(r + 1) * 4096u + (uint32_t)lane * 16u, logits);
            wait_async_le8();
        } else {
            wait_async_0();
        }

        const int rc = clampRow(r);
        const int t  = tgt[rc];              // uniform per wave -> scalar load

        const float4* base = reinterpret_cast<const float4*>(&lds[wave][j & 1][0]);

        // Pass 1: load 32 values/lane from LDS, row max.
        float4 v[8];
        float m = -INFINITY;
        #pragma unroll
        for (int k = 0; k < 8; ++k) {
            v[k] = base[32 * k + lane];
            m = fmaxf(m, fmaxf(fmaxf(v[k].x, v[k].y), fmaxf(v[k].z, v[k].w)));
        }
        m = wred_max(m);

        // Iteration k covers classes [128k,128k+128) = level-1 block k.
        // Lane group (lane>>2) within iteration covers one level-2 16-class block.
        const int i1 = t >> 7;          // which iteration holds the target block
        const int g2 = (t >> 4) & 7;    // lane group of target's level-2 block
        const int lt = (t >> 2) & 31;   // lane holding target element
        const int ct = t & 3;           // component within that lane's float4

        float z = 0.0f, s1 = 0.0f, s2 = 0.0f, tv = 0.0f;
        #pragma unroll
        for (int k = 0; k < 8; ++k) {
            const float ex = __expf(v[k].x - m);
            const float ey = __expf(v[k].y - m);
            const float ez = __expf(v[k].z - m);
            const float ew = __expf(v[k].w - m);
            const float s4 = (ex + ey) + (ez + ew);
            z += s4;
            if (k == i1) {
                s1 = s4;
                s2 = ((lane >> 2) == g2) ? s4 : 0.0f;
                if (lane == lt) {
                    const float fv = (ct == 0) ? v[k].x : (ct == 1) ? v[k].y
                                   : (ct == 2) ? v[k].z : v[k].w;
                    tv = fv - m;
                }
            }
        }
        z  = wred_sum(z);
        s1 = wred_sum(s1);
        s2 = wred_sum(s2);
        tv = wred_sum(tv);   // only one lane contributed; sum == broadcast

        // loss = lam0*logZ + (lam1-lam0)*logS1 + (lam2-lam1)*logS2 - lam2*t
        const float loss = lam0 * __logf(z)
                         + (lam1 - lam0) * __logf(s1)
                         + (lam2 - lam1) * __logf(s2)
                         - lam2 * tv;
        if (r < Btot) wloss += loss;
    }

    if (lane == 0) wsum[wave] = wloss;
    __syncthreads();
    if (threadIdx.x == 0) {
        float s = 0.0f;
        #pragma unroll
        for (int w = 0; w < WAVES_PER_BLOCK; ++w) s += wsum[w];
        partial[blockIdx.x] = s;
    }
}

__global__ __launch_bounds__(256) void hce_reduce_kernel(
        const float* __restrict__ partial, int n, float invB, float* __restrict__ out) {
    __shared__ float s[256];
    float v = 0.0f;
    for (int i = threadIdx.x; i < n; i += 256) v += partial[i];
    s[threadIdx.x] = v;
    __syncthreads();
    #pragma unroll
    for (int st = 128; st > 0; st >>= 1) {
        if ((int)threadIdx.x < st) s[threadIdx.x] += s[threadIdx.x + st];
        __syncthreads();
    }
    if (threadIdx.x == 0) out[0] = s[0] * invB;
}

extern "C" void kernel_launch(void* const* d_in, const int* in_sizes, int n_in,
                              void* d_out, int out_size, void* d_ws, size_t ws_size,
                              hipStream_t stream) {
    // setup_inputs order: logits, target_classes, L, lam, paths
    const float* logits = (const float*)d_in[0];
    const int*   tgt    = (const int*)d_in[1];
    const float* lam    = (const float*)d_in[3];
    float* out     = (float*)d_out;
    float* partial = (float*)d_ws;

    const int Btot = in_sizes[0] / C_CLASSES;   // 32768
    const int grid = (Btot + ROWS_PER_BLOCK - 1) / ROWS_PER_BLOCK;  // 1024

    hce_rows_kernel<<<grid, 256, 0, stream>>>(logits, tgt, lam, partial, Btot);
    hce_reduce_kernel<<<1, 256, 0, stream>>>(partial, grid, 1.0f / (float)Btot, out);
}